// RegionLoss_81286551044227
// MI455X (gfx1250) — compile-verified
//
#include <hip/hip_runtime.h>

// ---------------- problem constants (from reference) ----------------
#define NB      32
#define NH      76
#define NW      76
#define NAANCH  5
#define NCCLS   24
#define MAXGT   50
#define CHANS   (NAANCH * (5 + NCCLS))   // 145
#define HW      (NH * NW)                // 5776
#define CELLS   (NAANCH * HW)            // 28880
#define THRESH  0.6f
#define NOOBJ   1.0f
#define OBJW    5.0f
#define WSB     640                      // per-batch float stride in workspace

typedef __attribute__((ext_vector_type(2))) float v2f;
typedef __attribute__((ext_vector_type(8))) float v8f;

__constant__ float c_aw[NAANCH] = {1.3221f, 3.19275f, 5.05587f, 9.47112f, 11.2364f};
__constant__ float c_ah[NAANCH] = {1.73145f, 4.00944f, 8.09892f, 4.84053f, 10.0071f};

static __device__ __forceinline__ float sigmoidf(float v) {
    return 1.0f / (1.0f + expf(-v));
}

static __device__ __forceinline__ float iou_cwh(float x1, float y1, float w1, float h1,
                                                float x2, float y2, float w2, float h2) {
    float mx = fminf(x1 - 0.5f * w1, x2 - 0.5f * w2);
    float Mx = fmaxf(x1 + 0.5f * w1, x2 + 0.5f * w2);
    float my = fminf(y1 - 0.5f * h1, y2 - 0.5f * h2);
    float My = fmaxf(y1 + 0.5f * h1, y2 + 0.5f * h2);
    float cw = w1 + w2 - (Mx - mx);
    float ch = h1 + h2 - (My - my);
    float inter = ((cw > 0.0f) && (ch > 0.0f)) ? cw * ch : 0.0f;
    return inter / (w1 * h1 + w2 * h2 - inter);
}

static __device__ __forceinline__ float smooth_l1(float d) {
    float ad = fabsf(d);
    return (ad < 1.0f) ? 0.5f * d * d : ad - 0.5f;
}

// ---------------- kernel A: per-batch GT preprocessing ----------------
// One block per batch. Builds, per GT t:
//   gx,gy,gw,gh, valid flag  (for the max-IoU pass)
//   cell index (best_n*HW + gj*NW + gi, or -1), tx,ty,tw,th,tconf,tcls (scatter table)
__global__ void region_prep_kernel(const float* __restrict__ out,
                                   const float* __restrict__ tgt,
                                   float* __restrict__ ws,
                                   float* __restrict__ loss_out) {
    const int b   = blockIdx.x;
    const int tid = threadIdx.x;

    __shared__ float st[MAXGT * 5];
    __shared__ int   sval[MAXGT];

    for (int k = tid; k < MAXGT * 5; k += blockDim.x)
        st[k] = tgt[(size_t)b * MAXGT * 5 + k];
    __syncthreads();

    if (tid == 0) {
        if (b == 0) loss_out[0] = 0.0f;        // harness poisons d_out; we accumulate into it
        int prev = 1;                           // valid = cumprod(tgt[...,1] != 0)
        for (int t = 0; t < MAXGT; ++t) {
            int v = prev && (st[t * 5 + 1] != 0.0f);
            sval[t] = v;
            prev = v;
        }
    }
    __syncthreads();

    float* W = ws + (size_t)b * WSB;
    if (tid < MAXGT) {
        const int t = tid;
        const int v = sval[t];
        float gx = st[t * 5 + 1] * (float)NW;
        float gy = st[t * 5 + 2] * (float)NH;
        float gw = st[t * 5 + 3] * (float)NW;
        float gh = st[t * 5 + 4] * (float)NH;
        float cls = st[t * 5 + 0];

        // anchor argmax (first max wins, matching jnp.argmax)
        int bn = 0; float best = -1.0f;
        for (int a = 0; a < NAANCH; ++a) {
            float inter = fminf(c_aw[a], gw) * fminf(c_ah[a], gh);
            float aiou  = inter / (c_aw[a] * c_ah[a] + gw * gh - inter);
            if (aiou > best) { best = aiou; bn = a; }
        }

        int gi = min(max((int)gx, 0), NW - 1);
        int gj = min(max((int)gy, 0), NH - 1);
        float txv = gx - (float)gi;
        float tyv = gy - (float)gj;
        float twv = logf(fmaxf(gw, 1e-12f) / c_aw[bn]);
        float thv = logf(fmaxf(gh, 1e-12f) / c_ah[bn]);

        float tconf = 0.0f;
        int   cell  = -1;
        if (v) {
            size_t base = (((size_t)b * CHANS + bn * (5 + NCCLS)) * NH + gj) * NW + gi;
            float o0 = out[base];
            float o1 = out[base + (size_t)HW];
            float o2 = out[base + (size_t)2 * HW];
            float o3 = out[base + (size_t)3 * HW];
            float pbx = sigmoidf(o0) + (float)gi;
            float pby = sigmoidf(o1) + (float)gj;
            float pbw = expf(o2) * c_aw[bn];
            float pbh = expf(o3) * c_ah[bn];
            tconf = iou_cwh(gx, gy, gw, gh, pbx, pby, pbw, pbh);
            cell  = bn * HW + gj * NW + gi;
        }

        W[0 * MAXGT + t] = gx;
        W[1 * MAXGT + t] = gy;
        W[2 * MAXGT + t] = gw;
        W[3 * MAXGT + t] = gh;
        W[4 * MAXGT + t] = v ? 1.0f : 0.0f;
        ((int*)W)[5 * MAXGT + t] = cell;
        W[6 * MAXGT + t]  = txv;
        W[7 * MAXGT + t]  = tyv;
        W[8 * MAXGT + t]  = twv;
        W[9 * MAXGT + t]  = thv;
        W[10 * MAXGT + t] = tconf;
        W[11 * MAXGT + t] = cls;
    }
}

// ---------------- kernel B: main per-cell loss + WMMA wave reduction ----------------
__global__ void region_loss_kernel(const float* __restrict__ out,
                                   const float* __restrict__ ws,
                                   float* __restrict__ loss_out) {
    const int b    = blockIdx.y;
    const int tid  = threadIdx.x;
    const int cell = blockIdx.x * blockDim.x + tid;

    __shared__ float sgx[MAXGT], sgy[MAXGT], sgw[MAXGT], sgh[MAXGT], svld[MAXGT];
    __shared__ int   scell[MAXGT];
    __shared__ float stx[MAXGT], sty[MAXGT], stw[MAXGT], sth[MAXGT], stc[MAXGT], scls[MAXGT];
    __shared__ float swave[8];

    const float* W = ws + (size_t)b * WSB;
    if (tid < MAXGT) {
        sgx[tid]  = W[0 * MAXGT + tid];
        sgy[tid]  = W[1 * MAXGT + tid];
        sgw[tid]  = W[2 * MAXGT + tid];
        sgh[tid]  = W[3 * MAXGT + tid];
        svld[tid] = W[4 * MAXGT + tid];
        scell[tid] = ((const int*)W)[5 * MAXGT + tid];
        stx[tid]  = W[6 * MAXGT + tid];
        sty[tid]  = W[7 * MAXGT + tid];
        stw[tid]  = W[8 * MAXGT + tid];
        sth[tid]  = W[9 * MAXGT + tid];
        stc[tid]  = W[10 * MAXGT + tid];
        scls[tid] = W[11 * MAXGT + tid];
    }
    __syncthreads();

    const bool active = (cell < CELLS);
    const int c2 = active ? cell : 0;           // clamp so loads stay in-bounds
    const int a   = c2 / HW;
    const int rem = c2 - a * HW;
    const int j   = rem / NW;
    const int i   = rem - j * NW;

    const size_t base = (((size_t)b * CHANS + a * (5 + NCCLS)) * NH + j) * NW + i;
    float o0 = out[base];
    float o1 = out[base + (size_t)HW];
    float o2 = out[base + (size_t)2 * HW];
    float o3 = out[base + (size_t)3 * HW];
    float o4 = out[base + (size_t)4 * HW];

    float x = sigmoidf(o0);
    float y = sigmoidf(o1);
    float w = o2;
    float h = o3;
    float conf = sigmoidf(o4);

    float pbx = x + (float)i;
    float pby = y + (float)j;
    float pbw = expf(w) * c_aw[a];
    float pbh = expf(h) * c_ah[a];

    // max IoU vs valid GTs (validity is a prefix -> uniform break)
    float maxiou = 0.0f;
    for (int t = 0; t < MAXGT; ++t) {
        if (svld[t] == 0.0f) break;
        float iou = iou_cwh(pbx, pby, pbw, pbh, sgx[t], sgy[t], sgw[t], sgh[t]);
        maxiou = fmaxf(maxiou, iou);
    }

    // last-writer-wins scatter lookup (matches lax.scan semantics)
    int mt = -1;
    for (int t = 0; t < MAXGT; ++t)
        if (scell[t] == c2) mt = t;

    float conf_m, tx, ty, tw, th, tconf;
    int clsm = 0, tclsI = 0;
    if (mt >= 0) {
        conf_m = OBJW;
        tx = stx[mt]; ty = sty[mt]; tw = stw[mt]; th = sth[mt]; tconf = stc[mt];
        clsm = 1; tclsI = (int)scls[mt];
    } else {
        conf_m = (maxiou > THRESH) ? 0.0f : NOOBJ;
        tx = 0.5f; ty = 0.5f; tw = 0.0f; th = 0.0f; tconf = 0.0f;
    }

    float loss = 0.5f * (smooth_l1(x - tx) + smooth_l1(y - ty) +
                         smooth_l1(w - tw) + smooth_l1(h - th));
    float dc = conf - tconf;
    loss += 0.5f * conf_m * dc * dc;

    // focal class loss: only matched cells (<= ~50 per batch) ever contribute,
    // so we skip reading the 120 class channels everywhere else (89 MB saved).
    if (clsm) {
        float m  = -1e30f;
        float lt = 0.0f;
        for (int c = 0; c < NCCLS; ++c) {
            float lg = out[base + (size_t)(5 + c) * HW];
            m = fmaxf(m, lg);
            if (c == tclsI) lt = lg;
        }
        float s = 0.0f;
        for (int c = 0; c < NCCLS; ++c)
            s += expf(out[base + (size_t)(5 + c) * HW] - m);
        float lpt = (lt - m) - logf(s);
        float pt  = expf(lpt);
        float om  = 1.0f - pt;
        loss += -(om * om) * lpt;
    }

    if (!active) loss = 0.0f;

    // ---- wave32 reduction via V_WMMA_F32_16X16X4_F32: D = A(16x4) * ones(4x16) ----
    // A layout: lanes 0-15 supply K=0 (vgpr0) / K=1 (vgpr1) for row M=lane;
    // lanes 16-31 supply K=2/K=3 for row M=lane-16. With av = {loss, 0}, row-sum m =
    // loss[m] + loss[m+16]. D vgpr r holds rows r (lanes 0-15) / 8+r (lanes 16-31),
    // so summing the 8 D regs gives half-wave sums; shfl_xor(16) completes the wave sum.
    v2f av; av[0] = loss; av[1] = 0.0f;
    v2f bv; bv[0] = 1.0f; bv[1] = 1.0f;
    v8f cv = {};
    cv = __builtin_amdgcn_wmma_f32_16x16x4_f32(false, av, false, bv,
                                               (short)0, cv, false, false);
    float half = cv[0] + cv[1] + cv[2] + cv[3] + cv[4] + cv[5] + cv[6] + cv[7];
    float wsum = half + __shfl_xor(half, 16, 32);

    const int wave = tid >> 5;
    if ((tid & 31) == 0) swave[wave] = wsum;
    __syncthreads();
    if (tid == 0) {
        float tot = 0.0f;
        int nwaves = (blockDim.x + 31) >> 5;
        for (int k = 0; k < nwaves; ++k) tot += swave[k];
        atomicAdd(loss_out, tot);
    }
}

// ---------------- host launcher ----------------
extern "C" void kernel_launch(void* const* d_in, const int* in_sizes, int n_in,
                              void* d_out, int out_size, void* d_ws, size_t ws_size,
                              hipStream_t stream) {
    (void)in_sizes; (void)n_in; (void)out_size; (void)ws_size;
    const float* out = (const float*)d_in[0];   // (32, 145, 76, 76) fp32
    const float* tgt = (const float*)d_in[1];   // (32, 250) fp32
    float* ws   = (float*)d_ws;
    float* loss = (float*)d_out;

    region_prep_kernel<<<dim3(NB), dim3(64), 0, stream>>>(out, tgt, ws, loss);

    dim3 grid((CELLS + 255) / 256, NB);
    region_loss_kernel<<<grid, dim3(256), 0, stream>>>(out, ws, loss);
}